// MeanP2CPDistance_4939212390980
// MI455X (gfx1250) — compile-verified
//
#include <hip/hip_runtime.h>
#include <hip/hip_bf16.h>
#include <math.h>

typedef float v2f __attribute__((ext_vector_type(2)));
typedef float v8f __attribute__((ext_vector_type(8)));

#define MT 2  // row-tiles (of 16 source points) per wave

// One direction of the chamfer distance: for each point p in P (per batch),
// find min_q ||p-q|| over Q of the same batch, via WMMA-evaluated squared
// distances:  d2 = (-2px)*qx + (-2py)*qy + (px^2+py^2)*1 + 1*(qx^2+qy^2)
// Each wave handles MT 16-row tiles of P and streams all of Q.
// Writes one partial (sum of 16*MT sqrt-min distances) per wave.
__global__ __launch_bounds__(256) void chamfer_dir_kernel(
    const float* __restrict__ P,   // (B, nP, 2)
    const float* __restrict__ Q,   // (B, nQ, 2)
    float* __restrict__ partial,   // one float per wave
    int nP, int nQ)
{
    const int lane   = threadIdx.x & 31;
    const int waveId = (int)((blockIdx.x * blockDim.x + threadIdx.x) >> 5);
    const int tilesPerBatch = nP / (16 * MT);
    const int b    = waveId / tilesPerBatch;
    const int tile = waveId % tilesPerBatch;

    const int sub = lane & 15;   // row/col within 16-tile
    const int hi  = lane >> 4;   // 0: K0/K1 half, 1: K2/K3 half

    const float2* Pb = (const float2*)P + (size_t)b * nP;
    const float2* Qb = (const float2*)Q + (size_t)b * nQ;

    // Build A matrices: lane<16 -> (-2px, -2py); lane>=16 -> (|p|^2, 1)
    v2f a[MT];
#pragma unroll
    for (int mt = 0; mt < MT; ++mt) {
        const float2 p = Pb[(tile * MT + mt) * 16 + sub];
        v2f lo; lo.x = -2.0f * p.x;         lo.y = -2.0f * p.y;
        v2f hv; hv.x = p.x * p.x + p.y * p.y; hv.y = 1.0f;
        a[mt] = hi ? hv : lo;
    }

    float mins[MT][8];
#pragma unroll
    for (int mt = 0; mt < MT; ++mt)
#pragma unroll
        for (int r = 0; r < 8; ++r) mins[mt][r] = 3.0e38f;

    const int qTiles = nQ >> 4;
#pragma unroll 2
    for (int t = 0; t < qTiles; ++t) {
        const float2 q = Qb[t * 16 + sub];
        // B matrix: lane<16 -> (qx, qy); lane>=16 -> (1, |q|^2)
        v2f blo; blo.x = q.x;  blo.y = q.y;
        v2f bhi; bhi.x = 1.0f; bhi.y = q.x * q.x + q.y * q.y;
        const v2f bv = hi ? bhi : blo;
        const v8f c = {0.f, 0.f, 0.f, 0.f, 0.f, 0.f, 0.f, 0.f};
#pragma unroll
        for (int mt = 0; mt < MT; ++mt) {
            // D[r][lane] = squared distance of (row r+8*hi) to (col lane&15)
            v8f d = __builtin_amdgcn_wmma_f32_16x16x4_f32(
                false, a[mt], false, bv, (short)0, c, false, false);
#pragma unroll
            for (int r = 0; r < 8; ++r)
                mins[mt][r] = fminf(mins[mt][r], d[r]);
        }
    }

    // Cross-lane min within each 16-lane half, then sum sqrt over rows.
    float s = 0.0f;
#pragma unroll
    for (int mt = 0; mt < MT; ++mt) {
#pragma unroll
        for (int r = 0; r < 8; ++r) {
            float m = mins[mt][r];
            m = fminf(m, __shfl_xor(m, 1, 32));
            m = fminf(m, __shfl_xor(m, 2, 32));
            m = fminf(m, __shfl_xor(m, 4, 32));
            m = fminf(m, __shfl_xor(m, 8, 32));
            s += sqrtf(fmaxf(m, 0.0f));   // clamp tiny negative d2
        }
    }
    s += __shfl_xor(s, 16, 32);           // combine rows 0-7 and 8-15 halves
    if (lane == 0) partial[waveId] = s;
}

// Deterministic fixed-order reduction of both partial arrays -> scalar.
__global__ __launch_bounds__(256) void chamfer_reduce_kernel(
    const float* __restrict__ pu, int nu,
    const float* __restrict__ pv, int nv,
    float* __restrict__ out, float su_scale, float sv_scale)
{
    __shared__ float lu[256];
    __shared__ float lv[256];
    float su = 0.0f, sv = 0.0f;
    for (int i = threadIdx.x; i < nu; i += 256) su += pu[i];
    for (int i = threadIdx.x; i < nv; i += 256) sv += pv[i];
    lu[threadIdx.x] = su;
    lv[threadIdx.x] = sv;
    __syncthreads();
    for (int off = 128; off > 0; off >>= 1) {
        if ((int)threadIdx.x < off) {
            lu[threadIdx.x] += lu[threadIdx.x + off];
            lv[threadIdx.x] += lv[threadIdx.x + off];
        }
        __syncthreads();
    }
    if (threadIdx.x == 0)
        out[0] = lu[0] * su_scale + lv[0] * sv_scale;
}

extern "C" void kernel_launch(void* const* d_in, const int* in_sizes, int n_in,
                              void* d_out, int out_size, void* d_ws, size_t ws_size,
                              hipStream_t stream) {
    const float* u = (const float*)d_in[0];  // (B, N, 2)
    const float* v = (const float*)d_in[1];  // (B, M, 2)
    float* out = (float*)d_out;

    const int B = 32;
    const int N = in_sizes[0] / (B * 2);
    const int M = in_sizes[1] / (B * 2);

    const int wavesU = B * (N / (16 * MT));  // u -> nearest v
    const int wavesV = B * (M / (16 * MT));  // v -> nearest u

    float* pu = (float*)d_ws;
    float* pv = pu + wavesU;

    dim3 blk(256);
    chamfer_dir_kernel<<<dim3((wavesU * 32) / 256), blk, 0, stream>>>(u, v, pu, N, M);
    chamfer_dir_kernel<<<dim3((wavesV * 32) / 256), blk, 0, stream>>>(v, u, pv, M, N);

    const float su_scale = 1.0f / (2.0f * (float)B * (float)N);
    const float sv_scale = 1.0f / (2.0f * (float)B * (float)M);
    chamfer_reduce_kernel<<<dim3(1), blk, 0, stream>>>(pu, wavesU, pv, wavesV,
                                                       out, su_scale, sv_scale);
}